// AdaptiveFCN_84799834292726
// MI455X (gfx1250) — compile-verified
//
#include <hip/hip_runtime.h>

typedef float v2f __attribute__((ext_vector_type(2)));
typedef float v8f __attribute__((ext_vector_type(8)));
typedef long long lint;

#define B_   4
#define N_   2048
#define K_   20
#define BN_  8192            // B*N
#define MK_  163840LL        // B*N*K
#define EPS_ 1e-5f
#define NEG_ 0.2f

// ---------------------------------------------------------------- pts: (B,N,3) -> (3, B*N)
__global__ void k_pts(const float* __restrict__ x, float* __restrict__ P) {
    int t = blockIdx.x * blockDim.x + threadIdx.x;
    if (t >= BN_) return;
#pragma unroll
    for (int c = 0; c < 3; ++c) P[c * BN_ + t] = x[t * 3 + c];
}

// ---------------------------------------------------------------- KNN top-20 by d = 2<xi,xm> - |xi|^2 - |xm|^2
__global__ void k_knn(const float* __restrict__ P, int* __restrict__ idx) {
    __shared__ float sx[N_], sy[N_], sz[N_], sxx[N_];
    int bb  = blockIdx.x / (N_ / 256);
    int seg = blockIdx.x % (N_ / 256);
    for (int m = threadIdx.x; m < N_; m += 256) {
        float a = P[0 * BN_ + bb * N_ + m];
        float b = P[1 * BN_ + bb * N_ + m];
        float c = P[2 * BN_ + bb * N_ + m];
        sx[m] = a; sy[m] = b; sz[m] = c; sxx[m] = a * a + b * b + c * c;
    }
    __syncthreads();
    int n = seg * 256 + threadIdx.x;
    float xi = sx[n], yi = sy[n], zi = sz[n], xxn = sxx[n];
    float vals[K_]; int ids[K_];
#pragma unroll
    for (int k = 0; k < K_; ++k) { vals[k] = -3.4e38f; ids[k] = 0; }
    for (int m = 0; m < N_; ++m) {
        float d = 2.f * (xi * sx[m] + yi * sy[m] + zi * sz[m]) - xxn - sxx[m];
        if (d > vals[K_ - 1]) {                       // strict >  => stable (lowest index first on ties)
            int p = K_ - 1;
            while (p > 0 && d > vals[p - 1]) { vals[p] = vals[p - 1]; ids[p] = ids[p - 1]; --p; }
            vals[p] = d; ids[p] = m;
        }
    }
    lint base = (lint)(bb * N_ + n) * K_;
    for (int k = 0; k < K_; ++k) idx[base + k] = ids[k];
}

// ---------------------------------------------------------------- graph feature: Y[0:C]=nb-ctr, Y[C:2C]=ctr ; (2C x MK)
__global__ void k_ybuild(const float* __restrict__ F, const int* __restrict__ idx,
                         float* __restrict__ Y, int C) {
    lint m = (lint)blockIdx.x * blockDim.x + threadIdx.x;
    if (m >= MK_) return;
    lint nn = m / K_;                 // b*N + n
    int  b  = (int)(nn / N_);
    int  j  = idx[m];
    lint nb = (lint)b * N_ + j;
    for (int c = 0; c < C; ++c) {
        float fc = F[(lint)c * BN_ + nn];
        float fj = F[(lint)c * BN_ + nb];
        Y[(lint)c * MK_ + m]       = fj - fc;
        Y[(lint)(C + c) * MK_ + m] = fc;
    }
}

// ---------------------------------------------------------------- pad weight (OxCin) -> (OxCinP) with zeros
__global__ void k_padW(const float* __restrict__ Wsrc, float* __restrict__ Wdst,
                       int O, int Cin, int CinP) {
    int t = blockIdx.x * blockDim.x + threadIdx.x;
    if (t >= O * CinP) return;
    int r = t / CinP, c = t % CinP;
    Wdst[t] = (c < Cin) ? Wsrc[r * Cin + c] : 0.f;
}

// ---------------------------------------------------------------- WMMA fp32 GEMM: C(OxM) = W(OxCin) * act(X(CinxM))
// optional per-row transform of X: v = lrelu(v*xscale[k]+xshift[k])  (fused BatchNorm + leaky relu)
// wave tile: 64(O) x 32(M); 16x16x4 f32 WMMA; O % 64 == 0, Cin % 4 == 0, M % 32 == 0.
__global__ void __launch_bounds__(128)
k_gemm(const float* __restrict__ W, const float* __restrict__ X, float* __restrict__ Cm,
       int O, int Cin, lint M,
       const float* __restrict__ xscale, const float* __restrict__ xshift, int xlrelu) {
    int lane = threadIdx.x & 31;
    int wave = threadIdx.x >> 5;
    lint gw  = (lint)blockIdx.x * 4 + wave;
    int numOT = O >> 6;
    lint mT = gw / numOT;
    int  oT = (int)(gw % numOT);
    if (mT * 32 >= M) return;                     // whole-wave uniform exit (WMMA needs full EXEC)
    int half = lane >> 4;                         // K-half select
    int lr   = lane & 15;
    int kOff = half * 2;
    lint colBase = mT * 32;

    v8f acc[4][2];
#pragma unroll
    for (int i = 0; i < 4; ++i)
#pragma unroll
        for (int j = 0; j < 2; ++j)
#pragma unroll
            for (int q = 0; q < 8; ++q) acc[i][j][q] = 0.f;

#pragma unroll 2
    for (int kk = 0; kk < Cin; kk += 4) {
        int k0 = kk + kOff;                       // Cin % 4 == 0 -> always in bounds
        if (kk + 4 < Cin)
            __builtin_prefetch(&X[(lint)(kk + 4 + kOff) * M + colBase + lr], 0, 1);
        v2f a[4];
#pragma unroll
        for (int i = 0; i < 4; ++i) {
            lint row = (lint)(oT * 64 + i * 16 + lr);
            a[i] = *reinterpret_cast<const v2f*>(W + row * Cin + k0);   // 8B-aligned b64 load
        }
        v2f bv[2];
#pragma unroll
        for (int j = 0; j < 2; ++j) {
            lint col = colBase + j * 16 + lr;
            float b0 = X[(lint)k0 * M + col];
            float b1 = X[(lint)(k0 + 1) * M + col];
            if (xscale) {
                b0 = b0 * xscale[k0] + xshift[k0];
                b1 = b1 * xscale[k0 + 1] + xshift[k0 + 1];
                if (xlrelu) {
                    b0 = (b0 < 0.f) ? b0 * NEG_ : b0;
                    b1 = (b1 < 0.f) ? b1 * NEG_ : b1;
                }
            }
            bv[j].x = b0; bv[j].y = b1;
        }
#pragma unroll
        for (int i = 0; i < 4; ++i)
#pragma unroll
            for (int j = 0; j < 2; ++j)
                acc[i][j] = __builtin_amdgcn_wmma_f32_16x16x4_f32(
                    false, a[i], false, bv[j], (short)0, acc[i][j], false, false);
    }
#pragma unroll
    for (int i = 0; i < 4; ++i)
#pragma unroll
        for (int j = 0; j < 2; ++j) {
            lint col = colBase + j * 16 + lr;
#pragma unroll
            for (int v = 0; v < 8; ++v) {
                int row = oT * 64 + i * 16 + v + 8 * half;   // C/D layout: vgpr v -> M=v (+8 for hi lanes)
                Cm[(lint)row * M + col] = acc[i][j][v];
            }
        }
}

// ---------------------------------------------------------------- per-channel sum / sumsq
__global__ void k_zero(float* __restrict__ p, int n) {
    int t = blockIdx.x * blockDim.x + threadIdx.x;
    if (t < n) p[t] = 0.f;
}
__global__ void k_stats(const float* __restrict__ T, lint M, float* __restrict__ sums) {
    __shared__ float s1[256], s2[256];
    int c = blockIdx.y;
    const float* row = T + (lint)c * M;
    float a = 0.f, b = 0.f;
    for (lint i = (lint)blockIdx.x * 256 + threadIdx.x; i < M; i += (lint)gridDim.x * 256) {
        float v = row[i]; a += v; b += v * v;
    }
    s1[threadIdx.x] = a; s2[threadIdx.x] = b;
    __syncthreads();
    for (int st = 128; st > 0; st >>= 1) {
        if ((int)threadIdx.x < st) { s1[threadIdx.x] += s1[threadIdx.x + st]; s2[threadIdx.x] += s2[threadIdx.x + st]; }
        __syncthreads();
    }
    if (threadIdx.x == 0) {
        atomicAdd(&sums[2 * c],     s1[0]);
        atomicAdd(&sums[2 * c + 1], s2[0]);
    }
}
__global__ void k_finstats(const float* __restrict__ sums, const float* __restrict__ g,
                           const float* __restrict__ b, float* __restrict__ scale,
                           float* __restrict__ shift, int C, float invM) {
    int c = blockIdx.x * blockDim.x + threadIdx.x;
    if (c >= C) return;
    float mean = sums[2 * c] * invM;
    float var  = sums[2 * c + 1] * invM - mean * mean;
    float inv  = rsqrtf(var + EPS_);
    float sc   = g[c] * inv;
    scale[c] = sc;
    shift[c] = b[c] - mean * sc;
}

// ---------------------------------------------------------------- out[h,m] = sum_{c<6} T1[h*6+c,m]*XG[c,m]
__global__ void k_contract6(const float* __restrict__ T1, const float* __restrict__ XG,
                            float* __restrict__ T2) {
    lint m = (lint)blockIdx.x * blockDim.x + threadIdx.x;
    if (m >= MK_) return;
    float g0 = XG[0 * MK_ + m], g1 = XG[1 * MK_ + m], g2 = XG[2 * MK_ + m];
    float g3 = XG[3 * MK_ + m], g4 = XG[4 * MK_ + m], g5 = XG[5 * MK_ + m];
    for (int h = 0; h < 64; ++h) {
        const float* base = T1 + (lint)h * 6 * MK_ + m;
        float s = base[0] * g0 + base[MK_] * g1 + base[2 * MK_] * g2
                + base[3 * MK_] * g3 + base[4 * MK_] * g4 + base[5 * MK_] * g5;
        T2[(lint)h * MK_ + m] = s;
    }
}

// ---------------------------------------------------------------- max_k lrelu(bn(lin)+bn(res)) -> H rows [hOff, hOff+outC)
__global__ void k_maxlrelu(const float* __restrict__ T3, const float* __restrict__ T4,
                           const float* __restrict__ scL, const float* __restrict__ shL,
                           const float* __restrict__ scR, const float* __restrict__ shR,
                           float* __restrict__ H, int outC, int hOff) {
    lint t = (lint)blockIdx.x * blockDim.x + threadIdx.x;
    if (t >= (lint)outC * BN_) return;
    int  o  = (int)(t / BN_);
    lint m2 = t % BN_;
    float sl = scL[o], hl = shL[o], sr = scR[o], hr = shR[o];
    const float* p3 = T3 + (lint)o * MK_ + m2 * K_;
    const float* p4 = T4 + (lint)o * MK_ + m2 * K_;
    float mx = -3.4e38f;
#pragma unroll
    for (int k = 0; k < K_; ++k) {
        float s = (p3[k] * sl + hl) + (p4[k] * sr + hr);
        s = (s >= 0.f) ? s : NEG_ * s;
        mx = fmaxf(mx, s);
    }
    H[(lint)(hOff + o) * BN_ + m2] = mx;
}

// ---------------------------------------------------------------- out (B,N,1024): [h | relu(bn(edge))] transposed
__global__ void k_final(const float* __restrict__ H, const float* __restrict__ E,
                        const float* __restrict__ scC, const float* __restrict__ shC,
                        float* __restrict__ out) {
    lint t = (lint)blockIdx.x * blockDim.x + threadIdx.x;
    if (t >= (lint)BN_ * 1024) return;
    int  c  = (int)(t % 1024);
    lint m2 = t / 1024;
    float v;
    if (c < 512) v = H[(lint)c * BN_ + m2];
    else {
        int cc = c - 512;
        v = E[(lint)cc * BN_ + m2] * scC[cc] + shC[cc];
        v = v > 0.f ? v : 0.f;
    }
    out[t] = v;
}

// ================================================================ host driver
struct Conv { const float *W0,*W1,*Wl,*Wr,*b0,*b1,*bl,*br,*g0,*g1,*gl,*gr; };

extern "C" void kernel_launch(void* const* d_in, const int* in_sizes, int n_in,
                              void* d_out, int out_size, void* d_ws, size_t ws_size,
                              hipStream_t stream) {
    (void)out_size; (void)ws_size;
    // ---- detect flattening order via unambiguous sizes (Wc=262144, c1.W0=384, x=24576)
    const float *x, *Wc, *bc, *gc;
    Conv cv[4];
    bool sortedKeys;
    int  convBase, xPos, wcPos, bcPos, gcPos;
    if (in_sizes[0] == 24576 && n_in > 1 && in_sizes[1] == 262144) {        // x, sorted(params)
        sortedKeys = true;  xPos = 0; wcPos = 1; bcPos = 2; convBase = 3; gcPos = 51;
    } else if (in_sizes[0] == 262144) {                                      // sorted(params), x
        sortedKeys = true;  wcPos = 0; bcPos = 1; convBase = 2; gcPos = 50; xPos = 51;
    } else if (in_sizes[0] == 24576) {                                       // x, insertion(params)
        sortedKeys = false; xPos = 0; convBase = 1; wcPos = 49; gcPos = 50; bcPos = 51;
    } else {                                                                 // insertion(params), x
        sortedKeys = false; convBase = 0; wcPos = 48; gcPos = 49; bcPos = 50; xPos = 51;
    }
    x  = (const float*)d_in[xPos];
    Wc = (const float*)d_in[wcPos];
    bc = (const float*)d_in[bcPos];
    gc = (const float*)d_in[gcPos];
    for (int l = 0; l < 4; ++l) {
        const int b = convBase + 12 * l;
        if (sortedKeys) { // W0,W1,Wl,Wr,b0,b1,bl,br,g0,g1,gl,gr
            cv[l] = Conv{ (const float*)d_in[b+0],(const float*)d_in[b+1],(const float*)d_in[b+2],(const float*)d_in[b+3],
                          (const float*)d_in[b+4],(const float*)d_in[b+5],(const float*)d_in[b+6],(const float*)d_in[b+7],
                          (const float*)d_in[b+8],(const float*)d_in[b+9],(const float*)d_in[b+10],(const float*)d_in[b+11] };
        } else {          // W0,W1,Wl,Wr,g0,b0,g1,b1,gl,bl,gr,br
            cv[l] = Conv{ (const float*)d_in[b+0],(const float*)d_in[b+1],(const float*)d_in[b+2],(const float*)d_in[b+3],
                          (const float*)d_in[b+5],(const float*)d_in[b+7],(const float*)d_in[b+9],(const float*)d_in[b+11],
                          (const float*)d_in[b+4],(const float*)d_in[b+6],(const float*)d_in[b+8],(const float*)d_in[b+10] };
        }
    }

    // ---- workspace carve (aliasing: T2<-T0, T3<-T1) : ~670 MB
    char* w = (char*)d_ws; lint off = 0;
    auto carve = [&](lint bytes) -> void* { void* p = w + off; off = (off + bytes + 255) & ~255LL; return p; };
    float* P    = (float*)carve(3LL  * BN_ * 4);
    int*   IDX  = (int*)  carve(MK_ * 4);
    float* XG   = (float*)carve(8LL  * MK_ * 4);   // 6 used + 2 zero-pad rows (Cin=8 for layer 1)
    float* Y    = (float*)carve(256LL * MK_ * 4);
    float* T0   = (float*)carve(64LL  * MK_ * 4);
    float* T1   = (float*)carve(384LL * MK_ * 4);
    float* T4   = (float*)carve(256LL * MK_ * 4);
    float* H    = (float*)carve(512LL * BN_ * 4);
    float* E    = (float*)carve(512LL * BN_ * 4);
    float* SUMS = (float*)carve(1024 * 4);
    float* SC0  = (float*)carve(512 * 4);
    float* SH0  = (float*)carve(512 * 4);
    float* SCA  = (float*)carve(512 * 4);
    float* SHA  = (float*)carve(512 * 4);
    float* SCB  = (float*)carve(512 * 4);
    float* SHB  = (float*)carve(512 * 4);
    float* W0P  = (float*)carve(64 * 8 * 4);       // layer-1 W0 zero-padded to 64x8
    float* WRP  = (float*)carve(64 * 8 * 4);       // layer-1 Wr zero-padded to 64x8
    float* T2 = T0;   // alias (T0 dead when T2 produced)
    float* T3 = T1;   // alias (T1 dead when T3 produced)

    auto gemm = [&](const float* Wm, const float* X, float* Cm, int O, int Cin, lint M,
                    const float* xs, const float* xh, int xlr) {
        lint waves  = (lint)(O / 64) * (M / 32);
        lint blocks = (waves + 3) / 4;
        k_gemm<<<dim3((unsigned)blocks), dim3(128), 0, stream>>>(Wm, X, Cm, O, Cin, M, xs, xh, xlr);
    };
    auto stats = [&](const float* T, int C, lint M, const float* g, const float* b,
                     float* sc, float* sh) {
        k_zero<<<dim3((2 * C + 255) / 256), 256, 0, stream>>>(SUMS, 2 * C);
        k_stats<<<dim3(64, C), 256, 0, stream>>>(T, M, SUMS);
        k_finstats<<<dim3((C + 63) / 64), 64, 0, stream>>>(SUMS, g, b, sc, sh, C, 1.f / (float)M);
    };

    // ---- pipeline
    k_pts<<<dim3(BN_ / 256), 256, 0, stream>>>(x, P);
    k_knn<<<dim3(B_ * (N_ / 256)), 256, 0, stream>>>(P, IDX);
    k_ybuild<<<dim3((unsigned)(MK_ / 256)), 256, 0, stream>>>(P, IDX, XG, 3);   // xg == layer-1 Y (rows 0..5)
    k_zero<<<dim3((unsigned)((2 * MK_ + 255) / 256)), 256, 0, stream>>>(XG + 6 * MK_, (int)(2 * MK_));
    k_padW<<<dim3((64 * 8 + 255) / 256), 256, 0, stream>>>(cv[0].W0, W0P, 64, 6, 8);
    k_padW<<<dim3((64 * 8 + 255) / 256), 256, 0, stream>>>(cv[0].Wr, WRP, 64, 6, 8);

    const int   Cfeat[4] = {3, 64, 64, 128};
    const int   outC[4]  = {64, 64, 128, 256};
    const int   hOff[4]  = {0, 64, 128, 256};
    const float* Fs[4]   = {P, H, H + 64LL * BN_, H + 128LL * BN_};

    for (int l = 0; l < 4; ++l) {
        const int C = Cfeat[l];
        const float* Yl  = (l == 0) ? XG : Y;
        const int    CinY = (l == 0) ? 8 : 2 * C;          // layer 1 padded 6 -> 8
        const float* W0l = (l == 0) ? W0P : cv[l].W0;
        const float* Wrl = (l == 0) ? WRP : cv[l].Wr;
        if (l > 0) k_ybuild<<<dim3((unsigned)(MK_ / 256)), 256, 0, stream>>>(Fs[l], IDX, Y, C);
        gemm(W0l, Yl, T0, 64, CinY, MK_, nullptr, nullptr, 0);                  // ker pre-bn
        stats(T0, 64, MK_, cv[l].g0, cv[l].b0, SC0, SH0);
        gemm(cv[l].W1, T0, T1, 384, 64, MK_, SC0, SH0, 1);                      // W1 @ lrelu(bn(ker))
        k_contract6<<<dim3((unsigned)(MK_ / 256)), 256, 0, stream>>>(T1, XG, T2);
        stats(T2, 64, MK_, cv[l].g1, cv[l].b1, SC0, SH0);
        gemm(cv[l].Wl, T2, T3, outC[l], 64, MK_, SC0, SH0, 1);                  // lin pre-bn
        gemm(Wrl, Yl, T4, outC[l], CinY, MK_, nullptr, nullptr, 0);             // res pre-bn
        stats(T3, outC[l], MK_, cv[l].gl, cv[l].bl, SCA, SHA);
        stats(T4, outC[l], MK_, cv[l].gr, cv[l].br, SCB, SHB);
        lint tot = (lint)outC[l] * BN_;
        k_maxlrelu<<<dim3((unsigned)((tot + 255) / 256)), 256, 0, stream>>>(T3, T4, SCA, SHA, SCB, SHB, H, outC[l], hOff[l]);
    }

    gemm(Wc, H, E, 512, 512, (lint)BN_, nullptr, nullptr, 0);                   // edge pre-bn
    stats(E, 512, (lint)BN_, gc, bc, SC0, SH0);
    k_final<<<dim3((unsigned)(((lint)BN_ * 1024 + 255) / 256)), 256, 0, stream>>>(H, E, SC0, SH0, (float*)d_out);
}